// ProximalNeuralNetworkBlock_29377576305316
// MI455X (gfx1250) — compile-verified
//
#include <hip/hip_runtime.h>
#include <cstddef>

// ---------------------------------------------------------------------------
// f32 WMMA GEMM for gfx1250 (wave32).
// Block tile 128x128, BK=16, 128 threads = 4 wave32 in 2x2 grid.
// Each wave computes a 64x64 tile as 4x4 subtiles of 16x16 via
// V_WMMA_F32_16X16X4_F32 (exact f32 FMA semantics) -> 16 wmma per 8 LDS frag
// loads per K-step of 4.
// ---------------------------------------------------------------------------

typedef float v2f __attribute__((ext_vector_type(2)));
typedef float v8f __attribute__((ext_vector_type(8)));

#define BM 128
#define BN 128
#define BK 16
#define KPAD 4

enum {
  MODE_PLAIN  = 0,  // C = A*B
  MODE_ADD_I  = 1,  // C = A*B + I
  MODE_SCALE2 = 2,  // C = 2*(A*B)
  MODE_NEWTON = 3,  // C = 2*aux - A*B        (aux indexed like C, ld = ldc)
  MODE_TANH   = 4   // C = tanh(A*B + aux[n]) (aux = bias vector over columns)
};

// op(A)[m,k] = TA ? A[k*lda + m] : A[m*lda + k]
// op(B)[k,n] = TB ? B[n*ldb + k] : B[k*ldb + n]
// C[m*ldc + n] ; M = gridDim.y*128, N = gridDim.x*128 (all dims multiples of 128)
template <bool TA, bool TB, int MODE>
__global__ __launch_bounds__(128) void gemm_wmma_f32(
    float* __restrict__ C, const float* __restrict__ A,
    const float* __restrict__ B, const float* __restrict__ aux,
    int K, int lda, int ldb, int ldc)
{
  __shared__ float As[BM][BK + KPAD];
  __shared__ float Bs[BN][BK + KPAD];

  const int tid  = threadIdx.x;
  const int lane = tid & 31;
  const int wave = tid >> 5;   // 0..3
  const int wm   = wave & 1;   // 2 waves along M -> 64 rows each
  const int wn   = wave >> 1;  // 2 waves along N -> 64 cols each
  const int hi   = lane >> 4;  // lane half (K-pair select per ISA 7.12.2 layout)
  const int l15  = lane & 15;

  const int mBase = blockIdx.y * BM;
  const int nBase = blockIdx.x * BN;

  const v8f vzero = {0.f, 0.f, 0.f, 0.f, 0.f, 0.f, 0.f, 0.f};
  v8f acc[4][4];
#pragma unroll
  for (int i = 0; i < 4; ++i)
#pragma unroll
    for (int j = 0; j < 4; ++j) acc[i][j] = vzero;

  for (int k0 = 0; k0 < K; k0 += BK) {
    // ---- stage A tile (BMxBK) and B tile (BNxBK) into LDS.
    // 16 elements each per thread, mapped so every thread reads one
    // contiguous 64B run from global in all transpose variants.
#pragma unroll
    for (int j = 0; j < 16; ++j) {
      int e = tid * 16 + j;  // 0..2047
      int am, ak;
      if (TA) { ak = e >> 7; am = e & (BM - 1); }  // contiguous in m
      else    { am = e >> 4; ak = e & (BK - 1); }  // contiguous in k
      float va = TA ? A[(size_t)(k0 + ak) * lda + (mBase + am)]
                    : A[(size_t)(mBase + am) * lda + (k0 + ak)];
      As[am][ak] = va;

      int bn, bk;
      if (TB) { bn = e >> 4; bk = e & (BK - 1); }  // contiguous in k
      else    { bk = e >> 7; bn = e & (BN - 1); }  // contiguous in n
      float vb = TB ? B[(size_t)(nBase + bn) * ldb + (k0 + bk)]
                    : B[(size_t)(k0 + bk) * ldb + (nBase + bn)];
      Bs[bn][bk] = vb;
    }
    // prefetch next K-tile (global_prefetch_b8) to overlap with compute
    if (k0 + BK < K) {
      int e = tid * 16;
      int am, ak, bn, bk;
      if (TA) { ak = e >> 7; am = e & (BM - 1); }
      else    { am = e >> 4; ak = e & (BK - 1); }
      __builtin_prefetch(TA ? &A[(size_t)(k0 + BK + ak) * lda + (mBase + am)]
                            : &A[(size_t)(mBase + am) * lda + (k0 + BK + ak)], 0, 1);
      if (TB) { bn = e >> 4; bk = e & (BK - 1); }
      else    { bk = e >> 7; bn = e & (BN - 1); }
      __builtin_prefetch(TB ? &B[(size_t)(nBase + bn) * ldb + (k0 + BK + bk)]
                            : &B[(size_t)(k0 + BK + bk) * ldb + (nBase + bn)], 0, 1);
    }
    __syncthreads();

    // ---- 4 WMMA K-steps of 4; 16 v_wmma per step per wave
#pragma unroll
    for (int kk = 0; kk < BK; kk += 4) {
      v2f a[4], b[4];
#pragma unroll
      for (int im = 0; im < 4; ++im) {
        int m = wm * 64 + im * 16 + l15;
        a[im] = *(const v2f*)&As[m][kk + 2 * hi];  // K = kk+2*hi, kk+2*hi+1
      }
#pragma unroll
      for (int in = 0; in < 4; ++in) {
        int n = wn * 64 + in * 16 + l15;
        b[in] = *(const v2f*)&Bs[n][kk + 2 * hi];
      }
#pragma unroll
      for (int im = 0; im < 4; ++im)
#pragma unroll
        for (int in = 0; in < 4; ++in)
          acc[im][in] = __builtin_amdgcn_wmma_f32_16x16x4_f32(
              false, a[im], false, b[in], (short)0, acc[im][in], false, false);
    }
    __syncthreads();
  }

  // ---- epilogue: lane holds rows r+8*hi, col l15 of each 16x16 subtile
#pragma unroll
  for (int im = 0; im < 4; ++im) {
#pragma unroll
    for (int in = 0; in < 4; ++in) {
#pragma unroll
      for (int r = 0; r < 8; ++r) {
        int gm = mBase + wm * 64 + im * 16 + r + 8 * hi;
        int gn = nBase + wn * 64 + in * 16 + l15;
        float v = acc[im][in][r];
        if (MODE == MODE_ADD_I)       v += (gm == gn) ? 1.0f : 0.0f;
        else if (MODE == MODE_SCALE2) v *= 2.0f;
        else if (MODE == MODE_NEWTON) v = 2.0f * aux[(size_t)gm * ldc + gn] - v;
        else if (MODE == MODE_TANH)   v = tanhf(v + aux[gn]);
        C[(size_t)gm * ldc + gn] = v;
      }
    }
  }
}

// ---------------------------------------------------------------------------
// Small helper kernels for the Newton-Schulz initializer X0 = M / ||M||_inf
// ---------------------------------------------------------------------------
__global__ void zero_int_kernel(int* p) { *p = 0; }

__global__ __launch_bounds__(256) void row_abs_max_kernel(
    const float* __restrict__ Mm, int* __restrict__ normBits, int n)
{
  __shared__ float red[256];
  int row = blockIdx.x;
  float s = 0.f;
  for (int c = threadIdx.x; c < n; c += 256) s += fabsf(Mm[(size_t)row * n + c]);
  red[threadIdx.x] = s;
  __syncthreads();
  for (int off = 128; off > 0; off >>= 1) {
    if (threadIdx.x < off) red[threadIdx.x] += red[threadIdx.x + off];
    __syncthreads();
  }
  if (threadIdx.x == 0) atomicMax(normBits, __float_as_int(red[0]));  // positives: bit order == float order
}

__global__ __launch_bounds__(256) void scale_by_inv_kernel(
    float* __restrict__ X, const float* __restrict__ Mm,
    const int* __restrict__ normBits, int count)
{
  int i = blockIdx.x * 256 + threadIdx.x;
  float inv = 1.0f / __int_as_float(*normBits);
  if (i < count) X[i] = Mm[i] * inv;
}

// ---------------------------------------------------------------------------
// Host orchestration
// ---------------------------------------------------------------------------
extern "C" void kernel_launch(void* const* d_in, const int* in_sizes, int n_in,
                              void* d_out, int out_size, void* d_ws, size_t ws_size,
                              hipStream_t stream)
{
  const float* x    = (const float*)d_in[0];  // (8, 4096, 1024)
  const float* bias = (const float*)d_in[1];  // (4096,)
  const float* ttld = (const float*)d_in[2];  // (4096, 1024)
  float* out = (float*)d_out;                 // (8, 1024, 4096)

  const int I = 1024, H = 4096, NBATCH = 8;
  const int NEWTON_ITERS = 12;
  const size_t szY   = (size_t)H * I * sizeof(float);     // 16 MB
  const size_t szM   = (size_t)I * I * sizeof(float);     //  4 MB
  const size_t szAct = (size_t)H * 1024 * sizeof(float);  // 16 MB (one 1024-col panel)

  char* p = (char*)d_ws;
  float* Y0  = (float*)p; p += szY;
  float* Y1  = (float*)p; p += szY;
  float* Mm  = (float*)p; p += szM;
  float* X   = (float*)p; p += szM;
  float* X2  = (float*)p; p += szM;
  float* T   = (float*)p; p += szM;
  float* act = (float*)p; p += szAct;
  int* normBits = (int*)p;

  hipMemcpyAsync(Y0, ttld, szY, hipMemcpyDeviceToDevice, stream);

  dim3 blk(128);
  // ---- 5 Stiefel iterations: Y <- 2 * Y * inv(I + Y^T Y) ----
  for (int it = 0; it < 5; ++it) {
    // M = I + Y^T Y   (1024x1024, K=4096)
    gemm_wmma_f32<true, false, MODE_ADD_I>
        <<<dim3(I / BN, I / BM), blk, 0, stream>>>(Mm, Y0, Y0, nullptr, H, I, I, I);
    // ||M||_inf -> normBits ; X0 = M / ||M||_inf  (safe Newton-Schulz seed, M SPD)
    zero_int_kernel<<<1, 1, 0, stream>>>(normBits);
    row_abs_max_kernel<<<I, 256, 0, stream>>>(Mm, normBits, I);
    scale_by_inv_kernel<<<(I * I) / 256, 256, 0, stream>>>(X, Mm, normBits, I * I);
    // Newton-Schulz: X <- X (2I - M X) == 2X - X*(M*X)
    for (int j = 0; j < NEWTON_ITERS; ++j) {
      gemm_wmma_f32<false, false, MODE_PLAIN>
          <<<dim3(I / BN, I / BM), blk, 0, stream>>>(T, Mm, X, nullptr, I, I, I, I);
      gemm_wmma_f32<false, false, MODE_NEWTON>
          <<<dim3(I / BN, I / BM), blk, 0, stream>>>(X2, X, T, X, I, I, I, I);
      float* tmp = X; X = X2; X2 = tmp;
    }
    // Y_next = 2 * Y * X   (4096x1024, K=1024)
    gemm_wmma_f32<false, false, MODE_SCALE2>
        <<<dim3(I / BN, H / BM), blk, 0, stream>>>(Y1, Y0, X, nullptr, I, I, I, I);
    float* ty = Y0; Y0 = Y1; Y1 = ty;
  }
  // Y0 now holds mat (H x I)

  // ---- act = tanh(x_b @ mat^T + bias) ; out_b = mat^T @ act ----
  // Chunked over 1024-column panels of act to bound workspace (16 MB).
  const int NC = 1024;
  for (int b = 0; b < NBATCH; ++b) {
    const float* xb = x + (size_t)b * H * I;
    float* outb = out + (size_t)b * I * H;
    for (int kc = 0; kc < H; kc += NC) {
      // act[h, n] = tanh( sum_i xb[h,i] * mat[kc+n, i] + bias[kc+n] )
      gemm_wmma_f32<false, true, MODE_TANH>
          <<<dim3(NC / BN, H / BM), blk, 0, stream>>>(
              act, xb, Y0 + (size_t)kc * I, bias + kc, I, I, I, NC);
      // out[b, m, kc+n] = sum_h mat[h, m] * act[h, n]   (M=1024, N=NC, K=4096)
      gemm_wmma_f32<true, false, MODE_PLAIN>
          <<<dim3(NC / BN, I / BM), blk, 0, stream>>>(
              outb + kc, Y0, act, nullptr, H, I, NC, H);
    }
  }
}